// CapsuleLayer_2637109920060
// MI455X (gfx1250) — compile-verified
//
#include <hip/hip_runtime.h>
#include <math.h>

// Capsule routing for MI455X (gfx1250, wave32).
// W (32MB) + X (8MB) stay L2-resident; u_hat (512MB logical) is never
// materialized — recomputed per pass with V_WMMA_F32_16X16X4_F32.
// Transposed tiling: U^T[d,b] = W[n,m](16dx8i) x x^T(8ix16b), so each lane
// owns one batch column and 8 consecutive d's -> the d-reduction for routing
// logits needs only ONE ds_swizzle (SWAPX16) instead of a bpermute tree.

#define B_  128
#define M_  2048
#define I_  8
#define N_  32
#define D_  16
#define BT_ 8            // B_/16 b-tiles

typedef __attribute__((ext_vector_type(2))) float v2f;
typedef __attribute__((ext_vector_type(4))) float v4f;
typedef __attribute__((ext_vector_type(8))) float v8f;

__device__ __forceinline__ v8f wmma4(v2f a, v2f b, v8f c) {
  // 8 args: (neg_a, A, neg_b, B, c_mod, C, reuse_a, reuse_b)
  return __builtin_amdgcn_wmma_f32_16x16x4_f32(false, a, false, b, (short)0, c,
                                               false, false);
}

// A operand: W[n,m] as 16d x 8i (row d = lane&15).
// ISA 7.12.2 (32-bit A 16x4): lane<16 holds K=0,1 ; lane>=16 holds K=2,3.
__device__ __forceinline__ void load_w(const float* __restrict__ W, int n,
                                       int m, int col, int i0, v2f& a0, v2f& a1) {
  const float* wp = W + (size_t)n * (M_ * D_ * I_) + (size_t)m * (D_ * I_) +
                    col * I_;                       // row d=col, i contiguous
  __builtin_prefetch(wp + D_ * I_, 0, 0);           // next-m W line
  a0 = *(const v2f*)(wp + i0);                      // K = 0..3 slab
  a1 = *(const v2f*)(wp + 4 + i0);                  // K = 4..7 slab
}

// B operand: x^T as 8i x 16b (col b = lane&15), mirrored K layout.
__device__ __forceinline__ void load_x(const float* __restrict__ X, int b0,
                                       int m, int col, int i0, v2f& x0, v2f& x1) {
  const float* xp = X + (size_t)(b0 + col) * (M_ * I_) + (size_t)m * I_;
  x0 = *(const v2f*)(xp + i0);
  x1 = *(const v2f*)(xp + 4 + i0);
}

// beta[b] = sum_d U^T[d,b] * o[d]; lane holds d = j + 8*hi, so: 8 FMAs then
// one half-wave swap (ds_swizzle SWAPX16: xor=0x10, and=0x1f -> 0x401f).
__device__ __forceinline__ float beta_from(v8f U, v4f o0, v4f o1) {
  float t = U[0] * o0[0] + U[1] * o0[1] + U[2] * o0[2] + U[3] * o0[3] +
            U[4] * o1[0] + U[5] * o1[1] + U[6] * o1[2] + U[7] * o1[3];
  t += __int_as_float(__builtin_amdgcn_ds_swizzle(__float_as_int(t), 0x401f));
  return t;
}

// Pass 1 (iters 1,2): per (b,m) online softmax stats (max, sumexp) over n.
__global__ __launch_bounds__(256) void caps_softmax_stats(
    const float* __restrict__ X, const float* __restrict__ W,
    const float* __restrict__ Oacc, float* __restrict__ mx,
    float* __restrict__ se) {
  const int lane = threadIdx.x & 31;
  const int wave = threadIdx.x >> 5;
  const int col = lane & 15;
  const int hi = lane >> 4;
  const int i0 = 2 * hi;
  const int b0 = blockIdx.x * 16;
  const int b = b0 + col;
  const int MC = 16;                                 // m per wave
  const int m_base = (blockIdx.y * 8 + wave) * MC;

  for (int mi = 0; mi < MC; ++mi) {
    int m = m_base + mi;
    v2f x0, x1;
    load_x(X, b0, m, col, i0, x0, x1);               // shared across n
    float rmax = -3.0e38f, rsum = 0.0f;
    for (int n = 0; n < N_; ++n) {
      v2f a0, a1;
      load_w(W, n, m, col, i0, a0, a1);
      v8f U = {};
      U = wmma4(a0, x0, U);                          // K = 0..3
      U = wmma4(a1, x1, U);                          // K = 4..7
      const float* op = Oacc + ((size_t)b * N_ + n) * D_ + 8 * hi;
      v4f o0 = *(const v4f*)op;
      v4f o1 = *(const v4f*)(op + 4);
      float v = beta_from(U, o0, o1);
      float nm = fmaxf(rmax, v);                     // branch-free online softmax
      rsum = rsum * __expf(rmax - nm) + __expf(v - nm);
      rmax = nm;
    }
    if (hi == 0) {                                   // replicated across halves
      mx[(size_t)m * B_ + b] = rmax;
      se[(size_t)m * B_ + b] = rsum;
    }
  }
}

// Pass 2 (all iters): s[b,n,d] = sum_m c[b,n,m] * u_hat[b,n,m,d].
// One block owns (n, b-tile); each of its 8 waves sums 256 m and stores to the same s slot.
__global__ __launch_bounds__(256) void caps_accum_s(
    const float* __restrict__ X, const float* __restrict__ W,
    const float* __restrict__ Oacc, const float* __restrict__ mx,
    const float* __restrict__ se, float* __restrict__ s, int uniform_c) {
  const int lane = threadIdx.x & 31;
  const int wave = threadIdx.x >> 5;
  const int col = lane & 15;
  const int hi = lane >> 4;
  const int i0 = 2 * hi;
  const int n = blockIdx.x;
  const int b0 = blockIdx.y * 16;
  const int b = b0 + col;
  const int MC = M_ / 8;                             // 256 m per wave
  const int m_base = wave * MC;

  // O_acc slab for this (b, n): 8 consecutive d's, hoisted out of the m loop.
  const float* op = Oacc + ((size_t)b * N_ + n) * D_ + 8 * hi;
  v4f o0 = *(const v4f*)op;
  v4f o1 = *(const v4f*)(op + 4);

  v8f S = {};
  for (int mi = 0; mi < MC; ++mi) {
    int m = m_base + mi;
    v2f x0, x1, a0, a1;
    load_x(X, b0, m, col, i0, x0, x1);
    load_w(W, n, m, col, i0, a0, a1);
    v8f U = {};
    U = wmma4(a0, x0, U);
    U = wmma4(a1, x1, U);
    float c;
    if (uniform_c) {                                 // iter 0: softmax(0) = 1/N
      c = 1.0f / N_;
    } else {
      float bt = beta_from(U, o0, o1);
      c = __expf(bt - mx[(size_t)m * B_ + b]) / se[(size_t)m * B_ + b];
    }
#pragma unroll
    for (int j = 0; j < 8; ++j) S[j] += c * U[j];    // c uniform per lane
  }
  float* sp = s + ((size_t)b * N_ + n) * D_ + 8 * hi;
  v4f lo = {S[0], S[1], S[2], S[3]};
  v4f hi4 = {S[4], S[5], S[6], S[7]};
  *(v4f*)sp = lo;                                    // all 8 waves plain-store here (no reduction)
  *(v4f*)(sp + 4) = hi4;
}

// squash + output-accumulator update
__global__ __launch_bounds__(256) void caps_squash(const float* __restrict__ s,
                                                   float* __restrict__ Oacc,
                                                   float* __restrict__ out,
                                                   int add_oacc, int write_out) {
  int bn = blockIdx.x * blockDim.x + threadIdx.x;
  if (bn >= B_ * N_) return;
  const float* sp = s + (size_t)bn * D_;
  float s2 = 0.0f;
#pragma unroll
  for (int d = 0; d < D_; ++d) s2 += sp[d] * sp[d];
  float scale = s2 / (1.0f + s2) / sqrtf(s2 + 1e-7f);
#pragma unroll
  for (int d = 0; d < D_; ++d) {
    float o = scale * sp[d];
    if (add_oacc) Oacc[(size_t)bn * D_ + d] += o;
    if (write_out) out[(size_t)bn * D_ + d] = o;
  }
}

extern "C" void kernel_launch(void* const* d_in, const int* in_sizes, int n_in,
                              void* d_out, int out_size, void* d_ws,
                              size_t ws_size, hipStream_t stream) {
  const float* X = (const float*)d_in[0];            // [128,2048,8]
  const float* W = (const float*)d_in[1];            // [32,2048,16,8]
  float* out = (float*)d_out;                        // [128,32,16]
  float* ws = (float*)d_ws;

  float* s    = ws;                                  // 65536 f32
  float* Oacc = ws + 65536;                          // 65536 f32
  float* mx   = ws + 131072;                         // 262144 f32  [m][b]
  float* se   = ws + 131072 + 262144;                // 262144 f32  [m][b]

  hipMemsetAsync(Oacc, 0, 65536 * sizeof(float), stream);

  dim3 blk(256);
  dim3 gStats(BT_, M_ / (8 * 16));                   // 8 x 16 blocks
  dim3 gAcc(N_, BT_);                                // 32 x 8 blocks
  dim3 gSq((B_ * N_ + 255) / 256);

  for (int it = 0; it < 3; ++it) {
    if (it > 0)
      caps_softmax_stats<<<gStats, blk, 0, stream>>>(X, W, Oacc, mx, se);
    caps_accum_s<<<gAcc, blk, 0, stream>>>(X, W, Oacc, mx, se, s,
                                           it == 0 ? 1 : 0);
    int last = (it == 2);
    caps_squash<<<gSq, blk, 0, stream>>>(s, Oacc, out, last ? 0 : 1,
                                         last ? 1 : 0);
  }
}